// GraphAttention_48137993454074
// MI455X (gfx1250) — compile-verified
//
#include <hip/hip_runtime.h>
#include <hip/hip_bf16.h>
#include <math.h>

// ---------------------------------------------------------------------------
// Equivariant graph attention (e3nn-style) for MI455X / gfx1250, f32 WMMA.
//   M0=16 scalar, M1=8 vector inputs; Q0=8,Q1=4 keys; O0=16,O1=8 values.
//   Dominant FLOPs = per-edge MLPs -> batched GEMMs on V_WMMA_F32_16X16X4_F32.
// ---------------------------------------------------------------------------

typedef __attribute__((ext_vector_type(2))) float v2f;
typedef __attribute__((ext_vector_type(8))) float v8f;

__device__ __forceinline__ v8f wmma4(v2f a, v2f b, v8f c) {
  // D = A(16x4,f32) * B(4x16,f32) + C(16x16,f32)
  return __builtin_amdgcn_wmma_f32_16x16x4_f32(
      /*neg_a=*/false, a, /*neg_b=*/false, b,
      /*c_mod=*/(short)0, c, /*reuse_a=*/false, /*reuse_b=*/false);
}

#define INV_SQRT3  0.5773502691896258f
#define INV_SQRT2  0.7071067811865476f
#define INV_SQRT24 0.2041241452319315f   // 1/sqrt(m0+m1)
#define INV_SQRT32 0.17677669529663689f  // 1/sqrt(m0+2*m1) and 1/sqrt(HIDDEN)
#define INV_SQRT80 0.11180339887498949f  // 1/sqrt(Q0*Q0+Q1*Q1)

// ---------------------------------------------------------------------------
// fc_net (WMMA) + tensor-product consumption for one 16-edge group.
//   H = silu( ES[16x16] @ w1[16x32] * 0.25 )      (GEMM1, 8 wmma)
//   W = H @ w2[32 x w2cols] * 1/sqrt(32)          (GEMM2, 8 wmma per 16-col tile)
// Each 16x16 tile of W is staged in LDS and consumed by lanes 0..15
// (lane = edge) into acc_s / acc_v using the CG products.
// ---------------------------------------------------------------------------
__device__ __forceinline__ void fc_tp_pass(
    const float* __restrict__ es, const float* __restrict__ w1,
    const float* __restrict__ w2, int w2cols, int ntiles, bool vpath,
    int e0, int E, int tid, int half, int lane16,
    float Hsm[16][32], float Tsm[16][16],
    float xs_l[16][16], float xv_l[16][24],
    float pvv_l[16][8], float pxx_l[16][24],
    float shs_l[16], float shv_l[16][3],
    float acc_s[16][16], float acc_v[16][24]) {
  // ---- GEMM1: hidden activations ----
  int ea = e0 + lane16; if (ea >= E) ea = E - 1;
  #pragma unroll
  for (int nt = 0; nt < 2; ++nt) {
    v8f c = {0.f, 0.f, 0.f, 0.f, 0.f, 0.f, 0.f, 0.f};
    #pragma unroll
    for (int kt = 0; kt < 4; ++kt) {
      int kr = 4 * kt + 2 * half;
      v2f a, b;
      a.x = es[(size_t)ea * 16 + kr];           // A: lane16 = edge row
      a.y = es[(size_t)ea * 16 + kr + 1];
      b.x = w1[kr * 32 + nt * 16 + lane16];     // B: lane16 = out col
      b.y = w1[(kr + 1) * 32 + nt * 16 + lane16];
      c = wmma4(a, b, c);
    }
    #pragma unroll
    for (int r = 0; r < 8; ++r) {               // C/D: vgpr r -> row r+8*half
      float x = c[r] * 0.25f;                   // 1/sqrt(16)
      Hsm[8 * half + r][nt * 16 + lane16] = x / (1.f + __expf(-x));  // silu
    }
  }
  __syncthreads();
  // ---- re-layout H (C-layout) -> A operand registers ----
  v2f ah[8];
  #pragma unroll
  for (int kt = 0; kt < 8; ++kt) {
    ah[kt].x = Hsm[lane16][4 * kt + 2 * half];
    ah[kt].y = Hsm[lane16][4 * kt + 2 * half + 1];
  }
  __syncthreads();
  // ---- GEMM2 tiles, consumed immediately ----
  for (int ot = 0; ot < ntiles; ++ot) {
    v8f c = {0.f, 0.f, 0.f, 0.f, 0.f, 0.f, 0.f, 0.f};
    #pragma unroll
    for (int kt = 0; kt < 8; ++kt) {
      int kr = 4 * kt + 2 * half;
      v2f b;
      b.x = w2[kr * w2cols + ot * 16 + lane16];
      b.y = w2[(kr + 1) * w2cols + ot * 16 + lane16];
      c = wmma4(ah[kt], b, c);
    }
    #pragma unroll
    for (int r = 0; r < 8; ++r)
      Tsm[8 * half + r][lane16] = c[r] * INV_SQRT32;  // 1/sqrt(HIDDEN)
    __syncthreads();
    if (tid < 16) {
      const float shs = shs_l[tid];
      const float s0 = shv_l[tid][0], s1 = shv_l[tid][1], s2 = shv_l[tid][2];
      for (int n2 = 0; n2 < 16; ++n2) {
        int col = ot * 16 + n2;
        float w = Tsm[tid][n2];
        if (!vpath) {  // K weights: [16x8][8x8][16x4][8x4][8x4]
          if (col < 128)      { int i = col >> 3, o = col & 7;
                                acc_s[tid][o] += w * xs_l[tid][i] * shs; }
          else if (col < 192) { int t = col - 128, i = t >> 3, o = t & 7;
                                acc_s[tid][o] += w * pvv_l[tid][i]; }
          else if (col < 256) { int t = col - 192, i = t >> 2, o = t & 3;
                                float p = w * xs_l[tid][i];
                                acc_v[tid][o*3+0] += p * s0;
                                acc_v[tid][o*3+1] += p * s1;
                                acc_v[tid][o*3+2] += p * s2; }
          else if (col < 288) { int t = col - 256, i = t >> 2, o = t & 3;
                                float p = w * shs;
                                acc_v[tid][o*3+0] += p * xv_l[tid][i*3+0];
                                acc_v[tid][o*3+1] += p * xv_l[tid][i*3+1];
                                acc_v[tid][o*3+2] += p * xv_l[tid][i*3+2]; }
          else                { int t = col - 288, i = t >> 2, o = t & 3;
                                acc_v[tid][o*3+0] += w * pxx_l[tid][i*3+0];
                                acc_v[tid][o*3+1] += w * pxx_l[tid][i*3+1];
                                acc_v[tid][o*3+2] += w * pxx_l[tid][i*3+2]; }
        } else {       // V weights: [16x16][8x16][16x8][8x8][8x8]
          if (col < 256)      { int i = col >> 4, o = col & 15;
                                acc_s[tid][o] += w * xs_l[tid][i] * shs; }
          else if (col < 384) { int t = col - 256, i = t >> 4, o = t & 15;
                                acc_s[tid][o] += w * pvv_l[tid][i]; }
          else if (col < 512) { int t = col - 384, i = t >> 3, o = t & 7;
                                float p = w * xs_l[tid][i];
                                acc_v[tid][o*3+0] += p * s0;
                                acc_v[tid][o*3+1] += p * s1;
                                acc_v[tid][o*3+2] += p * s2; }
          else if (col < 576) { int t = col - 512, i = t >> 3, o = t & 7;
                                float p = w * shs;
                                acc_v[tid][o*3+0] += p * xv_l[tid][i*3+0];
                                acc_v[tid][o*3+1] += p * xv_l[tid][i*3+1];
                                acc_v[tid][o*3+2] += p * xv_l[tid][i*3+2]; }
          else                { int t = col - 576, i = t >> 3, o = t & 7;
                                acc_v[tid][o*3+0] += w * pxx_l[tid][i*3+0];
                                acc_v[tid][o*3+1] += w * pxx_l[tid][i*3+1];
                                acc_v[tid][o*3+2] += w * pxx_l[tid][i*3+2]; }
        }
      }
    }
    __syncthreads();
  }
}

// ---------------------------------------------------------------------------
// Kernels
// ---------------------------------------------------------------------------
__global__ void ga_zero_kernel(float* __restrict__ out, int n_out,
                               float* __restrict__ z, int n_z) {
  int i = blockIdx.x * blockDim.x + threadIdx.x;
  if (i < n_out) out[i] = 0.f;
  if (i < n_z) z[i] = 0.f;
}

__global__ void ga_q_kernel(const float* __restrict__ node_ft,
                            const float* __restrict__ wqs,
                            const float* __restrict__ wqv,
                            float* __restrict__ qws, int N) {
  int n = blockIdx.x * blockDim.x + threadIdx.x;
  if (n >= N) return;
  const float* x = node_ft + (size_t)n * 40;
  #pragma unroll
  for (int o = 0; o < 8; ++o) {
    float s = 0.f;
    #pragma unroll
    for (int i = 0; i < 16; ++i) s += x[i] * wqs[i * 8 + o];
    qws[(size_t)n * 20 + o] = s * 0.25f;  // 1/sqrt(16)
  }
  #pragma unroll
  for (int o = 0; o < 4; ++o)
    #pragma unroll
    for (int c = 0; c < 3; ++c) {
      float s = 0.f;
      #pragma unroll
      for (int i = 0; i < 8; ++i) s += x[16 + i * 3 + c] * wqv[i * 4 + o];
      qws[(size_t)n * 20 + 8 + o * 3 + c] = s * 0.35355339059327373f;  // 1/sqrt(8)
    }
}

__global__ __launch_bounds__(32) void ga_edge_kernel(
    const float* __restrict__ node_ft, const int* __restrict__ edge_index,
    const float* __restrict__ edge_sh, const float* __restrict__ edge_scalars,
    const float* __restrict__ fck_w1, const float* __restrict__ fck_w2,
    const float* __restrict__ fcv_w1, const float* __restrict__ fcv_w2,
    const float* __restrict__ wdot_s, const float* __restrict__ wdot_v,
    const float* __restrict__ qws, float* __restrict__ zws,
    float* __restrict__ exws, float* __restrict__ vbuf, int N, int E) {
  __shared__ float Hsm[16][32];
  __shared__ float Tsm[16][16];
  __shared__ float xs_l[16][16];
  __shared__ float xv_l[16][24];
  __shared__ float pvv_l[16][8];
  __shared__ float pxx_l[16][24];
  __shared__ float shs_l[16];
  __shared__ float shv_l[16][3];
  __shared__ float acc_s[16][16];
  __shared__ float acc_v[16][24];

  const int tid = threadIdx.x;
  const int half = tid >> 4;
  const int lane16 = tid & 15;
  const int e0 = blockIdx.x * 16;

  // ---- per-edge gather & CG-product precompute (lane = edge) ----
  if (tid < 16) {
    int e = e0 + tid; if (e >= E) e = E - 1;
    int snd = edge_index[e];                       // sender row
    float shs = edge_sh[(size_t)e * 4 + 0];
    float s0 = edge_sh[(size_t)e * 4 + 1];
    float s1 = edge_sh[(size_t)e * 4 + 2];
    float s2 = edge_sh[(size_t)e * 4 + 3];
    shs_l[tid] = shs;
    shv_l[tid][0] = s0; shv_l[tid][1] = s1; shv_l[tid][2] = s2;
    const float* nf = node_ft + (size_t)snd * 40;
    #pragma unroll
    for (int i = 0; i < 16; ++i) xs_l[tid][i] = nf[i];
    #pragma unroll
    for (int i = 0; i < 8; ++i) {
      float x = nf[16 + i * 3 + 0], y = nf[16 + i * 3 + 1], z = nf[16 + i * 3 + 2];
      xv_l[tid][i * 3 + 0] = x; xv_l[tid][i * 3 + 1] = y; xv_l[tid][i * 3 + 2] = z;
      pvv_l[tid][i] = (x * s0 + y * s1 + z * s2) * INV_SQRT3;
      pxx_l[tid][i * 3 + 0] = (y * s2 - z * s1) * INV_SQRT2;
      pxx_l[tid][i * 3 + 1] = (z * s0 - x * s2) * INV_SQRT2;
      pxx_l[tid][i * 3 + 2] = (x * s1 - y * s0) * INV_SQRT2;
    }
    #pragma unroll
    for (int o = 0; o < 16; ++o) acc_s[tid][o] = 0.f;
    #pragma unroll
    for (int j = 0; j < 24; ++j) acc_v[tid][j] = 0.f;
  }
  __syncthreads();

  // ---- K path: MLP (WMMA) + tensor product ----
  fc_tp_pass(edge_scalars, fck_w1, fck_w2, 320, 20, false, e0, E,
             tid, half, lane16, Hsm, Tsm, xs_l, xv_l, pvv_l, pxx_l,
             shs_l, shv_l, acc_s, acc_v);

  // ---- attention logit -> exp, segment-sum denominator ----
  if (tid < 16 && (e0 + tid) < E) {
    int e = e0 + tid;
    int recv = edge_index[E + e];                  // receiver row
    float ds = 0.f;
    for (int i = 0; i < 8; ++i) {
      float q = qws[(size_t)recv * 20 + i];
      for (int j = 0; j < 8; ++j) ds += q * acc_s[tid][j] * wdot_s[i * 8 + j];
    }
    float dv = 0.f;
    for (int i = 0; i < 4; ++i)
      for (int j = 0; j < 4; ++j) {
        float s = 0.f;
        for (int c = 0; c < 3; ++c)
          s += qws[(size_t)recv * 20 + 8 + i * 3 + c] * acc_v[tid][j * 3 + c];
        dv += s * wdot_v[i * 4 + j];
      }
    float dot = (ds * INV_SQRT24 + dv * INV_SQRT32 * INV_SQRT3) * INV_SQRT80;
    float ex = __expf(dot);
    exws[e] = ex;
    atomicAdd(&zws[recv], ex);
  }
  __syncthreads();
  if (tid < 16) {
    #pragma unroll
    for (int o = 0; o < 16; ++o) acc_s[tid][o] = 0.f;
    #pragma unroll
    for (int j = 0; j < 24; ++j) acc_v[tid][j] = 0.f;
  }
  __syncthreads();

  // ---- V path: MLP (WMMA) + tensor product ----
  fc_tp_pass(edge_scalars, fcv_w1, fcv_w2, 640, 40, true, e0, E,
             tid, half, lane16, Hsm, Tsm, xs_l, xv_l, pvv_l, pxx_l,
             shs_l, shv_l, acc_s, acc_v);

  if (tid < 16 && (e0 + tid) < E) {
    int e = e0 + tid;
    float* vb = vbuf + (size_t)e * 40;
    #pragma unroll
    for (int o = 0; o < 16; ++o) vb[o] = acc_s[tid][o] * INV_SQRT24;
    #pragma unroll
    for (int j = 0; j < 24; ++j) vb[16 + j] = acc_v[tid][j] * INV_SQRT32;
  }
}

__global__ void ga_scatter_kernel(const int* __restrict__ edge_index,
                                  const float* __restrict__ exws,
                                  const float* __restrict__ zws,
                                  const float* __restrict__ vbuf,
                                  float* __restrict__ out, int E) {
  int idx = blockIdx.x * blockDim.x + threadIdx.x;
  if (idx >= E * 40) return;
  int e = idx / 40, j = idx - e * 40;
  int recv = edge_index[E + e];
  float a = sqrtf(exws[e] / zws[recv]);            // sqrt(relu(alpha)), alpha>=0
  atomicAdd(&out[(size_t)recv * 40 + j], a * vbuf[(size_t)e * 40 + j]);
}

// ---------------------------------------------------------------------------
// Launch
// ---------------------------------------------------------------------------
extern "C" void kernel_launch(void* const* d_in, const int* in_sizes, int n_in,
                              void* d_out, int out_size, void* d_ws, size_t ws_size,
                              hipStream_t stream) {
  const float* node_ft      = (const float*)d_in[0];
  const int*   edge_index   = (const int*)d_in[1];
  const float* edge_sh      = (const float*)d_in[2];
  const float* edge_scalars = (const float*)d_in[3];
  const float* w_q_s        = (const float*)d_in[4];
  const float* w_q_v        = (const float*)d_in[5];
  const float* fck_w1       = (const float*)d_in[6];
  const float* fck_w2       = (const float*)d_in[7];
  const float* fcv_w1       = (const float*)d_in[8];
  const float* fcv_w2       = (const float*)d_in[9];
  const float* wdot_s       = (const float*)d_in[10];
  const float* wdot_v       = (const float*)d_in[11];
  float* out = (float*)d_out;

  const int N = in_sizes[0] / 40;     // node_ft [N,40]
  const int E = in_sizes[1] / 2;      // edge_index [2,E]

  // workspace layout (floats): q[N*20] | z[N] | ex[E] | vbuf[E*40] (~17.3 MB)
  float* qws  = (float*)d_ws;
  float* zws  = qws + (size_t)N * 20;
  float* exws = zws + N;
  float* vbuf = exws + E;

  int nz = N * 40;
  ga_zero_kernel<<<(nz + 255) / 256, 256, 0, stream>>>(out, nz, zws, N);
  ga_q_kernel<<<(N + 255) / 256, 256, 0, stream>>>(node_ft, w_q_s, w_q_v, qws, N);
  ga_edge_kernel<<<(E + 15) / 16, 32, 0, stream>>>(
      node_ft, edge_index, edge_sh, edge_scalars,
      fck_w1, fck_w2, fcv_w1, fcv_w2, wdot_s, wdot_v,
      qws, zws, exws, vbuf, N, E);
  int total = E * 40;
  ga_scatter_kernel<<<(total + 255) / 256, 256, 0, stream>>>(
      edge_index, exws, zws, vbuf, out, E);
}